// interaction_module_49031346651674
// MI455X (gfx1250) — compile-verified
//
#include <hip/hip_runtime.h>
#include <hip/hip_bf16.h>
#include <math.h>

typedef float v2f __attribute__((ext_vector_type(2)));
typedef float v8f __attribute__((ext_vector_type(8)));

#define N1   32768
#define N2   8192
#define KNN  8
#define MCH  32   // rel-MLP output channels
#define RCH  10   // relation feature length
#define FCH  20   // fc output channels
#define XCH  52   // fc input length (32 + 20)
#define NEG_SLOPE 0.2f
#define BN_EPS 1e-5f

// ---------------------------------------------------------------------------
// Kernel 1: brute-force KNN via V_WMMA_F32_16X16X4_F32 score tiles.
// score(p,q) = ||p||^2 - 2 p.q   (same argmin as ||p-q||^2)
//  A (16 cand x 4): [-2px, -2py, -2pz, ||p||^2]
//  B (4 x 16 qry):  [qx, qy, qz, 1]
// One wave handles one 16-query tile; 8 waves / block; 512 tiles total.
// Candidate loop unrolled x2 (two WMMA tiles / iter); tree-min screen so the
// expensive in-register insertion runs only when a tile actually improves
// some lane's 8th-best.
// ---------------------------------------------------------------------------
__global__ __launch_bounds__(256) void knn_kernel(const float* __restrict__ xyz1,
                                                  const float* __restrict__ xyz2,
                                                  int* __restrict__ idx_out) {
  __shared__ float lds_d[8][32][KNN];
  __shared__ int   lds_i[8][32][KNN];

  const int lane = threadIdx.x & 31;
  const int wave = threadIdx.x >> 5;
  const int tile = blockIdx.x * 8 + wave;   // 0..511
  const int qb   = tile * 16;
  const int m    = lane & 15;
  const int half = lane >> 4;

  // B operand (queries) -- loaded once, reused across the whole candidate loop
  const float qx = xyz2[(qb + m) * 3 + 0];
  const float qy = xyz2[(qb + m) * 3 + 1];
  const float qz = xyz2[(qb + m) * 3 + 2];
  v2f b;
  b.x = half ? qy : qx;      // VGPR0: lanes0-15 = K0 row (qx), lanes16-31 = K1 row (qy)
  b.y = half ? 1.0f : qz;    // VGPR1: lanes0-15 = K2 row (qz), lanes16-31 = K3 row (1)

  float bd[KNN];
  int   bi[KNN];
#pragma unroll
  for (int t = 0; t < KNN; ++t) { bd[t] = 3.0e38f; bi[t] = 0x7fffffff; }

  const float* pbase = xyz1 + (size_t)m * 3;

  for (int cb = 0; cb < N1; cb += 32) {
    const float* p0 = pbase + (size_t)cb * 3;
    const float* p1 = p0 + 16 * 3;
    __builtin_prefetch(p0 + 32 * 3, 0, 0);   // speculative global_prefetch_b8
    const float p0x = p0[0], p0y = p0[1], p0z = p0[2];
    const float p1x = p1[0], p1y = p1[1], p1z = p1[2];

    v2f a0, a1;
    a0.x = half ? (-2.0f * p0z) : (-2.0f * p0x);                     // K2 : K0
    a0.y = half ? (p0x * p0x + p0y * p0y + p0z * p0z) : (-2.0f * p0y); // K3 : K1
    a1.x = half ? (-2.0f * p1z) : (-2.0f * p1x);
    a1.y = half ? (p1x * p1x + p1y * p1y + p1z * p1z) : (-2.0f * p1y);

    v8f c0 = {}, c1 = {};
    c0 = __builtin_amdgcn_wmma_f32_16x16x4_f32(false, a0, false, b,
                                               (short)0, c0, false, false);
    c1 = __builtin_amdgcn_wmma_f32_16x16x4_f32(false, a1, false, b,
                                               (short)0, c1, false, false);

#pragma unroll
    for (int u = 0; u < 2; ++u) {
      const v8f c = u ? c1 : c0;
      const int base = cb + u * 16 + half * 8;

      // tree-min screen: skip the whole tile unless it can improve top-8
      const float m0 = fminf(fminf(c[0], c[1]), fminf(c[2], c[3]));
      const float m1 = fminf(fminf(c[4], c[5]), fminf(c[6], c[7]));
      if (fminf(m0, m1) < bd[KNN - 1]) {
#pragma unroll
        for (int i = 0; i < 8; ++i) {
          const float s  = c[i];
          const int   ci = base + i;
          if (s < bd[KNN - 1]) {
            bd[KNN - 1] = s; bi[KNN - 1] = ci;
#pragma unroll
            for (int j = KNN - 1; j > 0; --j) {
              if (bd[j] < bd[j - 1]) {
                float td = bd[j]; bd[j] = bd[j - 1]; bd[j - 1] = td;
                int   ti = bi[j]; bi[j] = bi[j - 1]; bi[j - 1] = ti;
              }
            }
          }
        }
      }
    }
  }

  // merge the two disjoint sorted-8 lists (lane l and lane l+16) via LDS
#pragma unroll
  for (int t = 0; t < KNN; ++t) { lds_d[wave][lane][t] = bd[t]; lds_i[wave][lane][t] = bi[t]; }
  __syncthreads();

  if (half == 0) {
    int pa = 0, pb = 0;
    int outIdx[KNN];
#pragma unroll
    for (int t = 0; t < KNN; ++t) {
      const float da = lds_d[wave][m][pa];
      const int   ia = lds_i[wave][m][pa];
      const float db = lds_d[wave][m + 16][pb];
      const int   ib = lds_i[wave][m + 16][pb];
      const bool takeA = (da < db) || (da == db && ia < ib);
      outIdx[t] = takeA ? ia : ib;
      if (takeA) ++pa; else ++pb;
    }
    const int firstIdx = outIdx[0];
#pragma unroll
    for (int t = 0; t < KNN; ++t) {
      int v = outIdx[t];
      if (v == 0) v = firstIdx;  // reference: idx = where(idx==0, idx[:, :1], idx)
      idx_out[(size_t)(qb + m) * KNN + t] = v;
    }
  }
}

// ---------------------------------------------------------------------------
// Kernel 2: per-wave partial sums (sum, sumsq) of h = relation . W_rel^T + b
// One thread per query n; loops k=0..7. Deterministic shuffle reductions.
// partials layout: [256 waves][64] (32 sum, 32 sumsq)
// ---------------------------------------------------------------------------
__global__ __launch_bounds__(256) void rel_stats_kernel(const float* __restrict__ xyz1,
                                                        const float* __restrict__ xyz2,
                                                        const int*   __restrict__ idxbuf,
                                                        const float* __restrict__ W_rel,
                                                        const float* __restrict__ b_rel,
                                                        float* __restrict__ partials) {
  __shared__ float sW[MCH * RCH];
  __shared__ float sB[MCH];
  for (int i = threadIdx.x; i < MCH * RCH; i += 256) sW[i] = W_rel[i];
  if (threadIdx.x < MCH) sB[threadIdx.x] = b_rel[threadIdx.x];
  __syncthreads();

  const int n = blockIdx.x * 256 + threadIdx.x;   // 0..8191
  const float cx = xyz2[n * 3 + 0];
  const float cy = xyz2[n * 3 + 1];
  const float cz = xyz2[n * 3 + 2];

  float sum[MCH], sq[MCH];
#pragma unroll
  for (int o = 0; o < MCH; ++o) { sum[o] = 0.0f; sq[o] = 0.0f; }

  for (int k = 0; k < KNN; ++k) {
    const int j = idxbuf[(size_t)n * KNN + k];
    const float gx = xyz1[(size_t)j * 3 + 0];
    const float gy = xyz1[(size_t)j * 3 + 1];
    const float gz = xyz1[(size_t)j * 3 + 2];
    const float ox = gx - cx, oy = gy - cy, oz = gz - cz;
    const float dist = sqrtf(ox * ox + oy * oy + oz * oz);
    const float rel[RCH] = {ox, oy, oz, cx, cy, cz, gx, gy, gz, dist};
#pragma unroll
    for (int o = 0; o < MCH; ++o) {
      float h = sB[o];
#pragma unroll
      for (int cc = 0; cc < RCH; ++cc) h += rel[cc] * sW[o * RCH + cc];
      sum[o] += h;
      sq[o]  += h * h;
    }
  }

#pragma unroll
  for (int o = 0; o < MCH; ++o) {
    float s = sum[o], q = sq[o];
    for (int off = 16; off >= 1; off >>= 1) {
      s += __shfl_xor(s, off, 32);
      q += __shfl_xor(q, off, 32);
    }
    sum[o] = s; sq[o] = q;
  }
  const int gwave = (blockIdx.x * 256 + threadIdx.x) >> 5;   // 0..255
  if ((threadIdx.x & 31) == 0) {
#pragma unroll
    for (int o = 0; o < MCH; ++o) {
      partials[(size_t)gwave * 64 + o]        = sum[o];
      partials[(size_t)gwave * 64 + MCH + o]  = sq[o];
    }
  }
}

// stats: [32 mean][32 rstd]
__global__ void rel_finalize_kernel(const float* __restrict__ partials,
                                    float* __restrict__ stats) {
  __shared__ float tot[64];
  const int t = threadIdx.x;   // 64 threads
  if (t < 64) {
    float s = 0.0f;
    for (int w = 0; w < 256; ++w) s += partials[(size_t)w * 64 + t];
    tot[t] = s;
  }
  __syncthreads();
  if (t < MCH) {
    const float inv = 1.0f / (float)(N2 * KNN);
    const float mean = tot[t] * inv;
    const float var  = tot[MCH + t] * inv - mean * mean;
    stats[t]        = mean;
    stats[MCH + t]  = rsqrtf(var + BN_EPS);
  }
}

// ---------------------------------------------------------------------------
// Kernel 4: recompute h, BN + leaky -> weights, weighted mean over k,
// fc matmul (52 -> 20), write y_pre, accumulate fc BN partials.
// partials2 layout: [256 waves][40] (20 sum, 20 sumsq)
// ---------------------------------------------------------------------------
__global__ __launch_bounds__(256) void fuse_kernel(const float* __restrict__ xyz1,
                                                   const float* __restrict__ feat1,
                                                   const float* __restrict__ xyz2,
                                                   const float* __restrict__ feat2,
                                                   const int*   __restrict__ idxbuf,
                                                   const float* __restrict__ W_rel,
                                                   const float* __restrict__ b_rel,
                                                   const float* __restrict__ g_rel,
                                                   const float* __restrict__ bt_rel,
                                                   const float* __restrict__ relstats,
                                                   const float* __restrict__ W_fc,
                                                   const float* __restrict__ b_fc,
                                                   float* __restrict__ y_pre,
                                                   float* __restrict__ partials2) {
  __shared__ float sW[MCH * RCH];
  __shared__ float sB[MCH], sMean[MCH], sRstd[MCH], sG[MCH], sBt[MCH];
  __shared__ float sWfc[FCH * XCH];
  __shared__ float sBfc[FCH];
  for (int i = threadIdx.x; i < MCH * RCH; i += 256) sW[i] = W_rel[i];
  for (int i = threadIdx.x; i < FCH * XCH; i += 256) sWfc[i] = W_fc[i];
  if (threadIdx.x < MCH) {
    sB[threadIdx.x]    = b_rel[threadIdx.x];
    sMean[threadIdx.x] = relstats[threadIdx.x];
    sRstd[threadIdx.x] = relstats[MCH + threadIdx.x];
    sG[threadIdx.x]    = g_rel[threadIdx.x];
    sBt[threadIdx.x]   = bt_rel[threadIdx.x];
  }
  if (threadIdx.x < FCH) sBfc[threadIdx.x] = b_fc[threadIdx.x];
  __syncthreads();

  const int n = blockIdx.x * 256 + threadIdx.x;   // 0..8191
  const float cx = xyz2[n * 3 + 0];
  const float cy = xyz2[n * 3 + 1];
  const float cz = xyz2[n * 3 + 2];

  float upd[MCH];
#pragma unroll
  for (int o = 0; o < MCH; ++o) upd[o] = 0.0f;

  for (int k = 0; k < KNN; ++k) {
    const int j = idxbuf[(size_t)n * KNN + k];
    const float gx = xyz1[(size_t)j * 3 + 0];
    const float gy = xyz1[(size_t)j * 3 + 1];
    const float gz = xyz1[(size_t)j * 3 + 2];
    const float ox = gx - cx, oy = gy - cy, oz = gz - cz;
    const float dist = sqrtf(ox * ox + oy * oy + oz * oz);
    const float rel[RCH] = {ox, oy, oz, cx, cy, cz, gx, gy, gz, dist};

    const float4* frow = (const float4*)(feat1 + (size_t)j * MCH);
    float ft[MCH];
#pragma unroll
    for (int v = 0; v < MCH / 4; ++v) {
      const float4 f4 = frow[v];
      ft[v * 4 + 0] = f4.x; ft[v * 4 + 1] = f4.y;
      ft[v * 4 + 2] = f4.z; ft[v * 4 + 3] = f4.w;
    }

#pragma unroll
    for (int o = 0; o < MCH; ++o) {
      float h = sB[o];
#pragma unroll
      for (int cc = 0; cc < RCH; ++cc) h += rel[cc] * sW[o * RCH + cc];
      const float hn = (h - sMean[o]) * sRstd[o] * sG[o] + sBt[o];
      const float w  = (hn >= 0.0f) ? hn : NEG_SLOPE * hn;
      upd[o] += ft[o] * w;
    }
  }
#pragma unroll
  for (int o = 0; o < MCH; ++o) upd[o] *= (1.0f / (float)KNN);

  float f2[FCH];
#pragma unroll
  for (int cc = 0; cc < FCH; ++cc) f2[cc] = feat2[(size_t)n * FCH + cc];

  const int gwave = (blockIdx.x * 256 + threadIdx.x) >> 5;   // 0..255
#pragma unroll
  for (int jj = 0; jj < FCH; ++jj) {
    float acc = sBfc[jj];
#pragma unroll
    for (int cc = 0; cc < MCH; ++cc) acc += upd[cc] * sWfc[jj * XCH + cc];
#pragma unroll
    for (int cc = 0; cc < FCH; ++cc) acc += f2[cc] * sWfc[jj * XCH + MCH + cc];
    y_pre[(size_t)n * FCH + jj] = acc;

    float s = acc, q = acc * acc;
    for (int off = 16; off >= 1; off >>= 1) {
      s += __shfl_xor(s, off, 32);
      q += __shfl_xor(q, off, 32);
    }
    if ((threadIdx.x & 31) == 0) {
      partials2[(size_t)gwave * 40 + jj]        = s;
      partials2[(size_t)gwave * 40 + FCH + jj]  = q;
    }
  }
}

// stats2: [20 mean][20 rstd]
__global__ void fc_finalize_kernel(const float* __restrict__ partials2,
                                   float* __restrict__ stats2) {
  __shared__ float tot[40];
  const int t = threadIdx.x;   // 64 threads, 40 used
  if (t < 40) {
    float s = 0.0f;
    for (int w = 0; w < 256; ++w) s += partials2[(size_t)w * 40 + t];
    tot[t] = s;
  }
  __syncthreads();
  if (t < FCH) {
    const float inv = 1.0f / (float)N2;
    const float mean = tot[t] * inv;
    const float var  = tot[FCH + t] * inv - mean * mean;
    stats2[t]        = mean;
    stats2[FCH + t]  = rsqrtf(var + BN_EPS);
  }
}

__global__ __launch_bounds__(256) void out_kernel(const float* __restrict__ y_pre,
                                                  const float* __restrict__ stats2,
                                                  const float* __restrict__ g_fc,
                                                  const float* __restrict__ bt_fc,
                                                  float* __restrict__ out) {
  const int i = blockIdx.x * 256 + threadIdx.x;
  if (i < N2 * FCH) {
    const int j = i % FCH;
    const float v = (y_pre[i] - stats2[j]) * stats2[FCH + j] * g_fc[j] + bt_fc[j];
    out[i] = (v >= 0.0f) ? v : NEG_SLOPE * v;
  }
}

// ---------------------------------------------------------------------------
extern "C" void kernel_launch(void* const* d_in, const int* in_sizes, int n_in,
                              void* d_out, int out_size, void* d_ws, size_t ws_size,
                              hipStream_t stream) {
  const float* xyz1   = (const float*)d_in[0];
  const float* feat1  = (const float*)d_in[1];
  const float* xyz2   = (const float*)d_in[2];
  const float* feat2  = (const float*)d_in[3];
  const float* W_rel  = (const float*)d_in[4];
  const float* b_rel  = (const float*)d_in[5];
  const float* g_rel  = (const float*)d_in[6];
  const float* bt_rel = (const float*)d_in[7];
  const float* W_fc   = (const float*)d_in[8];
  const float* b_fc   = (const float*)d_in[9];
  const float* g_fc   = (const float*)d_in[10];
  const float* bt_fc  = (const float*)d_in[11];
  float* out = (float*)d_out;

  // workspace layout (bytes), ~1 MB total; every word written before read.
  char* ws = (char*)d_ws;
  int*   idxbuf    = (int*)  (ws + 0);                        // 8192*8*4   = 262144
  float* relpart   = (float*)(ws + 262144);                   // 256*64*4   =  65536
  float* relstats  = (float*)(ws + 262144 + 65536);           // 64*4       =    256
  float* fcpart    = (float*)(ws + 262144 + 65536 + 256);     // 256*40*4   =  40960
  float* fcstats   = (float*)(ws + 262144 + 65536 + 256 + 40960);          // 40*4
  float* y_pre     = (float*)(ws + 262144 + 65536 + 256 + 40960 + 256);    // 8192*20*4

  knn_kernel<<<64, 256, 0, stream>>>(xyz1, xyz2, idxbuf);
  rel_stats_kernel<<<32, 256, 0, stream>>>(xyz1, xyz2, idxbuf, W_rel, b_rel, relpart);
  rel_finalize_kernel<<<1, 64, 0, stream>>>(relpart, relstats);
  fuse_kernel<<<32, 256, 0, stream>>>(xyz1, feat1, xyz2, feat2, idxbuf,
                                      W_rel, b_rel, g_rel, bt_rel, relstats,
                                      W_fc, b_fc, y_pre, fcpart);
  fc_finalize_kernel<<<1, 64, 0, stream>>>(fcpart, fcstats);
  out_kernel<<<(N2 * FCH + 255) / 256, 256, 0, stream>>>(y_pre, fcstats, g_fc, bt_fc, out);
}